// Demolition_splitweight_Conv2d_74234214744869
// MI455X (gfx1250) — compile-verified
//
#include <hip/hip_runtime.h>

typedef float v2f __attribute__((ext_vector_type(2)));
typedef float v8f __attribute__((ext_vector_type(8)));

#define C_IN   32
#define OUT_CH 32
#define HH     64
#define WW     64
#define NBITS  8

// Q_SCALE = (2^(5-1)-1)/1.2 = 12.5 ; dequant 1/12.5 folded into coefs below.
#define QSCALE 12.5f

// Block = 256 threads = 8 waves. Block handles one (b,h) row:
//   wave = mt*2 + jt, mt in [0,4) -> w tile of 16, jt in [0,2) -> j tile of 16.
// Each wave accumulates over all 32 input channels and 8 weight bits itself
// (quant + coef-weighted sums commute, so no cross-wave reduction / atomics).
//
// GEMM per (c, bit): D[16 spatial x 16 j] = A[16 x K12] * B[K12 x 16]
//   K 0..8  : 3x3 im2col taps / weights
//   K 9     : A = 1.0, B = bias  -> bias folded into the WMMA chain
//   K 10,11 : zero padding
__global__ __launch_bounds__(256)
void demolition_splitweight_conv_kernel(const float* __restrict__ x,       // [8,32,64,64]
                                        const float* __restrict__ weight,  // [8,1024,1,3,3]
                                        const float* __restrict__ bias,    // [8,1024]
                                        float* __restrict__ out)           // [8,32,64,64]
{
    __shared__ float xrow[3 * 66];   // 3 kernel rows, 66 cols (w = -1..64), zero padded

    const int tid  = threadIdx.x;
    const int lane = tid & 31;
    const int wid  = tid >> 5;       // 0..7
    const int mt   = wid >> 1;       // w tile index 0..3
    const int jt   = wid & 1;        // j tile index 0..1

    const int blk = blockIdx.x;      // b*64 + h
    const int b   = blk >> 6;
    const int h   = blk & 63;

    const int l16 = lane & 15;       // A: spatial row M ; B/D: column N (=j)
    const int hi  = lane >> 4;       // half-wave select
    const int w0  = mt * 16;
    const int jn  = jt * 16 + l16;   // this lane's j column

    // Per-lane LDS gather indices for A chunks 0/1 (K = 4*ch + 2*hi + v, all < 8)
    // and chunk 2 low half (K = 8, hi==0 only).
    int a_idx[5];
    int b_idx[5];
#pragma unroll
    for (int ch = 0; ch < 2; ++ch) {
#pragma unroll
        for (int v = 0; v < 2; ++v) {
            int k  = ch * 4 + hi * 2 + v;
            int kh = k / 3, kw = k % 3;
            a_idx[ch * 2 + v] = kh * 66 + (w0 + l16 + kw);
            b_idx[ch * 2 + v] = k;
        }
    }
    a_idx[4] = 2 * 66 + (w0 + l16 + 2);  // K=8 -> kh=2, kw=2
    b_idx[4] = 8;

    // coef[k] * (1/Q_SCALE): dequant scale folded into the bit coefficients.
    const float coefq[NBITS] = {
        (-128.0f / 127.0f) / QSCALE, (1.0f / 127.0f) / QSCALE,
        (2.0f / 127.0f) / QSCALE,    (4.0f / 127.0f) / QSCALE,
        (8.0f / 127.0f) / QSCALE,    (16.0f / 127.0f) / QSCALE,
        (32.0f / 127.0f) / QSCALE,   (64.0f / 127.0f) / QSCALE };

    const bool lo = (hi == 0);
    v8f acc = {0.f, 0.f, 0.f, 0.f, 0.f, 0.f, 0.f, 0.f};

    for (int c = 0; c < C_IN; ++c) {
        __syncthreads();  // protect LDS from previous iteration's readers
        // Stage padded x rows for (b, c, h-1..h+1, -1..64) into LDS.
        for (int i = tid; i < 3 * 66; i += 256) {
            int r  = i / 66;
            int wc = i % 66;
            int hs = h + r - 1;
            int ws = wc - 1;
            float v = 0.0f;
            if (hs >= 0 && hs < HH && ws >= 0 && ws < WW)
                v = x[(((size_t)b * C_IN + c) * HH + hs) * WW + ws];
            xrow[i] = v;
        }
        __syncthreads();

        // A operand chunks (shared across all 8 bit iterations).
        v2f a0, a1, a2;
        a0.x = xrow[a_idx[0]];
        a0.y = xrow[a_idx[1]];
        a1.x = xrow[a_idx[2]];
        a1.y = xrow[a_idx[3]];
        a2.x = lo ? xrow[a_idx[4]] : 0.0f;  // K=8 tap (hi==0), K=10 pad (hi==1)
        a2.y = lo ? 1.0f : 0.0f;            // K=9 bias lane,   K=11 pad

#pragma unroll
        for (int kb = 0; kb < NBITS; ++kb) {
            const size_t ocb = (size_t)kb * (C_IN * OUT_CH) + (size_t)c * OUT_CH + jn;
            const float* wp  = weight + ocb * 9;

            v2f b0, b1, b2;
            b0.x = wp[b_idx[0]];
            b0.y = wp[b_idx[1]];
            b1.x = wp[b_idx[2]];
            b1.y = wp[b_idx[3]];
            b2.x = lo ? wp[b_idx[4]] : 0.0f;   // K=8 weight / K=10 pad
            b2.y = lo ? bias[ocb] : 0.0f;      // K=9 bias   / K=11 pad

            // D = conv + bias over K=12 (bias folded), exact f32 accumulation.
            v8f d = {0.f, 0.f, 0.f, 0.f, 0.f, 0.f, 0.f, 0.f};
            d = __builtin_amdgcn_wmma_f32_16x16x4_f32(false, a0, false, b0, (short)0, d, false, false);
            d = __builtin_amdgcn_wmma_f32_16x16x4_f32(false, a1, false, b1, (short)0, d, false, false);
            d = __builtin_amdgcn_wmma_f32_16x16x4_f32(false, a2, false, b2, (short)0, d, false, false);

            // fake-quant (round-half-even via rintf/RNE); dequant+coef fused in ckq.
            const float ckq = coefq[kb];
#pragma unroll
            for (int r = 0; r < 8; ++r) {
                float q = rintf(d[r] * QSCALE);
                acc[r]  = fmaf(ckq, q, acc[r]);
            }
        }
    }

    // D layout: VGPR r -> row M = r + hi*8 (spatial w), lane column N = jn (j).
#pragma unroll
    for (int r = 0; r < 8; ++r) {
        int w = w0 + hi * 8 + r;
        out[(((size_t)b * OUT_CH + jn) * HH + h) * WW + w] = acc[r];
    }
}

extern "C" void kernel_launch(void* const* d_in, const int* in_sizes, int n_in,
                              void* d_out, int out_size, void* d_ws, size_t ws_size,
                              hipStream_t stream) {
    (void)in_sizes; (void)n_in; (void)d_ws; (void)ws_size; (void)out_size;
    const float* x      = (const float*)d_in[0];   // [8,32,64,64]
    const float* weight = (const float*)d_in[1];   // [8,1024,1,3,3]
    const float* bias   = (const float*)d_in[2];   // [8,1024]
    float* out          = (float*)d_out;           // [8,32,64,64]

    dim3 grid(8 * 64);   // one block per (b, h) row
    dim3 block(256);     // 8 waves: 4 w-tiles x 2 j-tiles
    demolition_splitweight_conv_kernel<<<grid, block, 0, stream>>>(x, weight, bias, out);
}